// CapsuleLayer_32959579029575
// MI455X (gfx1250) — compile-verified
//
#include <hip/hip_runtime.h>

// CDNA5 / gfx1250: wave32, WMMA f32 16x16x4, transposed tiling (C = [o x b]).
typedef __attribute__((ext_vector_type(2))) float v2f;
typedef __attribute__((ext_vector_type(8))) float v8f;

#define NCAPS   10
#define NROUTES 1152
#define CIN     16
#define COUT    32
#define BATCH   128
#define BTILE   16
#define WAVES   4
#define RPW     (NROUTES / WAVES)   // 288 routes per wave
#define LSTRIDE (NROUTES + 1)       // pad: 1152 % 64 == 0 would 16-way bank-conflict
#define RSTRIDE 17                  // padded batch stride for output reduce buffer
#define SMEM_FLOATS (BTILE*LSTRIDE + WAVES*COUT*RSTRIDE + WAVES*BTILE*2)

union V8 { v8f v; float f[8]; };

// One route: C[o,b] = sum_c W[c,o] * x[b,c]  (A = W^T 16x4 tiles, B = x 4x16 tiles).
// A f32 16x4 layout: vgpr0 = K+{0|2}, vgpr1 = K+{1|3} for lo|hi half; K-offset 2*hi is
// folded into wA (as +2*hi*COUT) and into xB (as +2*hi).  wA also carries +m16 (the o col).
__device__ __forceinline__ void priors_tile(const float* __restrict__ xB,
                                            const float* __restrict__ wA,
                                            V8& c0, V8& c1) {
  v8f acc0 = {}; v8f acc1 = {};
#pragma unroll
  for (int kb = 0; kb < 4; ++kb) {
    v2f b = *(const v2f*)(xB + 4*kb);                       // x[b, c..c+1], contiguous
    v2f aLo, aHi;
    aLo.x = wA[(4*kb+0)*COUT];       aLo.y = wA[(4*kb+1)*COUT];      // o in [0,16)
    aHi.x = wA[(4*kb+0)*COUT + 16];  aHi.y = wA[(4*kb+1)*COUT + 16]; // o in [16,32)
    acc0 = __builtin_amdgcn_wmma_f32_16x16x4_f32(false, aLo, false, b, (short)0, acc0, false, false);
    acc1 = __builtin_amdgcn_wmma_f32_16x16x4_f32(false, aHi, false, b, (short)0, acc1, false, false);
  }
  c0.v = acc0; c1.v = acc1;
}

__global__ __launch_bounds__(WAVES*32)
void caps_route_kernel(const float* __restrict__ x,
                       const float* __restrict__ W,
                       float* __restrict__ out) {
  extern __shared__ float smem[];
  float* l_lds = smem;                       // [BTILE][LSTRIDE] logits per (batch, route)
  float* redO  = smem + BTILE*LSTRIDE;       // [WAVES][COUT][RSTRIDE] output partials
  float* redMZ = redO + WAVES*COUT*RSTRIDE;  // [WAVES][BTILE][2] (max, Z) partials

  const int cap  = blockIdx.y;
  const int b0   = blockIdx.x * BTILE;
  const int wave = threadIdx.x >> 5;
  const int lane = threadIdx.x & 31;
  const int m16  = lane & 15;     // batch column owned by this lane
  const int hi   = lane >> 4;
  const int r0   = wave * RPW;

  // x as B operand: x[(b0+m16), r, c], K-offset 2*hi folded in.
  const float* xB0 = x + ((size_t)(b0 + m16) * NROUTES) * CIN + 2*hi;
  // W^T as A operand: W[cap, r, c, o], o = m16 (+16), K-offset 2*hi folded in (stride COUT).
  const float* wA0 = W + (size_t)cap * NROUTES * CIN * COUT + (2*hi)*COUT + m16;

  V8 oA0, oA1;        // outputs, C layout: o = g + 8*hi (+16 for oA1), b = m16
  float rowM, rowZinv; // per-lane softmax state for batch row b = m16

  // ---- accumulate pass: outputs = sum_r gc(p, ||u||^2) * priors[r,:] ----
  auto accum_pass = [&](bool uniform) {
#pragma unroll
    for (int g = 0; g < 8; ++g) { oA0.f[g] = 0.f; oA1.f[g] = 0.f; }
    for (int r = r0; r < r0 + RPW; ++r) {
      V8 c0, c1;
      priors_tile(xB0 + (size_t)r*CIN, wA0 + (size_t)r*CIN*COUT, c0, c1);
      float t = 0.f;
#pragma unroll
      for (int g = 0; g < 8; ++g) t += c0.f[g]*c0.f[g] + c1.f[g]*c1.f[g];
      float S = t + __shfl_xor(t, 16, 32);   // full sum over o (both halves)
      float p;
      if (uniform) {
        p = 1.0f / (float)NROUTES;           // softmax of all-zero logits
      } else {
        float l = l_lds[m16*LSTRIDE + r];
        p = __expf(l - rowM) * rowZinv;
      }
      float ps = p*p*S;
      // squash coefficient, branch-free: ps==0 when S==0 -> gc==0
      float gc = ps * __builtin_amdgcn_rcpf(1.f + ps) *
                 __builtin_amdgcn_rsqf(fmaxf(S, 1e-35f));
#pragma unroll
      for (int g = 0; g < 8; ++g) {
        oA0.f[g] += gc * c0.f[g];
        oA1.f[g] += gc * c1.f[g];
      }
    }
    // cross-wave reduction of the [32 x 16] output tile
#pragma unroll
    for (int g = 0; g < 8; ++g) {
      redO[(wave*COUT + g + 8*hi)*RSTRIDE + m16]      = oA0.f[g];
      redO[(wave*COUT + 16 + g + 8*hi)*RSTRIDE + m16] = oA1.f[g];
    }
    __syncthreads();
#pragma unroll
    for (int g = 0; g < 8; ++g) {
      float s0 = 0.f, s1 = 0.f;
#pragma unroll
      for (int w = 0; w < WAVES; ++w) {
        s0 += redO[(w*COUT + g + 8*hi)*RSTRIDE + m16];
        s1 += redO[(w*COUT + 16 + g + 8*hi)*RSTRIDE + m16];
      }
      oA0.f[g] = s0; oA1.f[g] = s1;
    }
    __syncthreads();
  };

  // ---- logits pass: l[b][r] (+)= priors[r,:] . outputs[b,:]; online max/Z ----
  auto logits_pass = [&](bool add) {
    float mRun = -3.0e38f, zRun = 0.f;
    for (int r = r0; r < r0 + RPW; ++r) {
      V8 c0, c1;
      priors_tile(xB0 + (size_t)r*CIN, wA0 + (size_t)r*CIN*COUT, c0, c1);
      float t = 0.f;
#pragma unroll
      for (int g = 0; g < 8; ++g) t += c0.f[g]*oA0.f[g] + c1.f[g]*oA1.f[g];
      float d = t + __shfl_xor(t, 16, 32);
      float l = add ? (l_lds[m16*LSTRIDE + r] + d) : d;
      if (hi == 0) l_lds[m16*LSTRIDE + r] = l;   // one writer per (b, r)
      float mo = mRun;
      float mn = fmaxf(mo, l);
      zRun = zRun * __expf(mo - mn) + __expf(l - mn);
      mRun = mn;
    }
    // cross-wave combine of (max, Z) with exp rescaling
    if (hi == 0) {
      redMZ[(wave*BTILE + m16)*2 + 0] = mRun;
      redMZ[(wave*BTILE + m16)*2 + 1] = zRun;
    }
    __syncthreads();
    float Mt = -3.0e38f;
#pragma unroll
    for (int w = 0; w < WAVES; ++w)
      Mt = fmaxf(Mt, redMZ[(w*BTILE + m16)*2 + 0]);
    float Zt = 0.f;
#pragma unroll
    for (int w = 0; w < WAVES; ++w)
      Zt += redMZ[(w*BTILE + m16)*2 + 1] * __expf(redMZ[(w*BTILE + m16)*2 + 0] - Mt);
    rowM = Mt;
    rowZinv = __builtin_amdgcn_rcpf(Zt);
    __syncthreads();
  };

  // 3 routing iterations; priors recomputed each pass (inputs are L2-resident)
  accum_pass(true);     // iter 0: uniform probs
  logits_pass(false);   // logits_1 = priors . out0
  accum_pass(false);    // iter 1
  logits_pass(true);    // logits_2 = logits_1 + priors . out1
  accum_pass(false);    // iter 2: final outputs

  // out[cap, b, 0, 0, o]; lane owns (b = m16, o = g + 8*hi (+16))
  if (wave == 0) {
#pragma unroll
    for (int g = 0; g < 8; ++g) {
      out[((size_t)cap*BATCH + b0 + m16)*COUT + g + 8*hi]      = oA0.f[g];
      out[((size_t)cap*BATCH + b0 + m16)*COUT + 16 + g + 8*hi] = oA1.f[g];
    }
  }
}

extern "C" void kernel_launch(void* const* d_in, const int* in_sizes, int n_in,
                              void* d_out, int out_size, void* d_ws, size_t ws_size,
                              hipStream_t stream) {
  (void)in_sizes; (void)n_in; (void)out_size; (void)d_ws; (void)ws_size;
  const float* x = (const float*)d_in[0];        // [128,1152,16] f32
  const float* W = (const float*)d_in[1];        // [10,1152,16,32] f32
  dim3 grid(BATCH / BTILE, NCAPS);               // (8, 10)
  caps_route_kernel<<<grid, WAVES*32, SMEM_FLOATS*sizeof(float), stream>>>(
      x, W, (float*)d_out);
}